// TemporalAttention_24008867185225
// MI455X (gfx1250) — compile-verified
//
#include <hip/hip_runtime.h>
#include <hip/hip_bf16.h>

// Problem dims (from reference): B=16, N=2048, F=32, T=288
#define B_ 16
#define N_ 2048
#define F_ 32
#define T_ 288
#define NCHUNK 64
#define NCHUNKS (N_ / NCHUNK)   // 32 chunks over N per batch

typedef __attribute__((ext_vector_type(2))) float v2f;
typedef __attribute__((ext_vector_type(8))) float v8f;

// ---------------------------------------------------------------------------
// f32 WMMA helper: D(16x16) += A(16x4) * B(4x16), all fp32.
// A-frag per lane: M = lane&15, K = (lane>>4)*2 + {0,1}  (2 VGPRs)
// B-frag per lane: N = lane&15, K = (lane>>4)*2 + {0,1}  (2 VGPRs)
// C/D per lane:    N = lane&15, M = vgpr + (lane>>4)*8   (8 VGPRs)
// ---------------------------------------------------------------------------
__device__ __forceinline__ v8f wmma_f32_k4(v2f a, v2f b, v8f c) {
    return __builtin_amdgcn_wmma_f32_16x16x4_f32(
        /*neg_a=*/false, a, /*neg_b=*/false, b,
        /*c_mod=*/(short)0, c, /*reuse_a=*/false, /*reuse_b=*/false);
}

// ---------------------------------------------------------------------------
// Kernel 1: single streaming pass over x (1.21 GB -> HBM bound, the whole cost).
// For each (b, n): thread t computes
//   C[b,n,t]      = sum_f x[b,n,f,t] * u3[f]
//   a[f] (regs)  += x[b,n,f,t] * u1[n]   (accumulated over an N-chunk of 64)
// Partial A written per chunk (deterministic; no fp atomics).
// ---------------------------------------------------------------------------
__global__ __launch_bounds__(T_) void k1_stream(
    const float* __restrict__ x, const float* __restrict__ u1,
    const float* __restrict__ u3, float* __restrict__ C,
    float* __restrict__ Apart) {
    const int t = threadIdx.x;       // 0..287 (coalesced along contiguous T dim)
    const int chunk = blockIdx.x;    // 0..31
    const int b = blockIdx.y;        // 0..15
    const int n0 = chunk * NCHUNK;

    float su3[F_];
#pragma unroll
    for (int f = 0; f < F_; ++f) su3[f] = u3[f];   // uniform -> scalar loads

    float a[F_];
#pragma unroll
    for (int f = 0; f < F_; ++f) a[f] = 0.f;

    for (int ni = 0; ni < NCHUNK; ++ni) {
        const int n = n0 + ni;
        const float w1 = u1[n];      // uniform -> scalar load
        const float* xp = x + ((size_t)(b * N_ + n) * F_) * T_ + t;
        float c = 0.f;
#pragma unroll
        for (int f = 0; f < F_; ++f) {
            float v = __builtin_nontemporal_load(xp + f * T_); // stream x, don't pollute L2
            c += v * su3[f];
            a[f] = fmaf(v, w1, a[f]);
        }
        C[(size_t)(b * N_ + n) * T_ + t] = c;      // 37.7 MB, stays hot in L2 for k3
    }

    float* ap = Apart + ((size_t)(b * NCHUNKS + chunk) * T_ + t) * F_;
#pragma unroll
    for (int f = 0; f < F_; ++f) ap[f] = a[f];
}

// ---------------------------------------------------------------------------
// Kernel 1b: deterministic reduction of Apart over the 32 chunks -> A[b,t,f].
// ---------------------------------------------------------------------------
__global__ __launch_bounds__(256) void k1b_reduceA(
    const float* __restrict__ Apart, float* __restrict__ A) {
    const int idx = blockIdx.x * 256 + threadIdx.x;   // over B_*T_*F_ = 147456
    if (idx >= B_ * T_ * F_) return;
    const int b = idx / (T_ * F_);
    const int tf = idx % (T_ * F_);
    const float* p = Apart + (size_t)b * NCHUNKS * T_ * F_ + tf;
    float s = 0.f;
#pragma unroll 8
    for (int c = 0; c < NCHUNKS; ++c) s += p[(size_t)c * T_ * F_];
    A[idx] = s;
}

// ---------------------------------------------------------------------------
// Kernel 2: D[b,t,n] = sum_f A[b,t,f] * u2[f,n]   (per b: 288x32 @ 32x2048)
// One wave per (b, t-tile16, group of 8 n-tiles); A-frags preloaded once (K=32).
// ---------------------------------------------------------------------------
__global__ __launch_bounds__(256) void k2_gemm(
    const float* __restrict__ A, const float* __restrict__ u2,
    float* __restrict__ D) {
    const int wave = (blockIdx.x * 256 + threadIdx.x) >> 5;  // 4608 waves
    const int lane = threadIdx.x & 31;
    const int ng = wave % 16;              // n-group: 8 tiles of 16 -> covers 2048
    const int tt = (wave / 16) % 18;       // t-tile of 16 -> covers 288
    const int b = wave / (16 * 18);
    const int m = lane & 15;
    const int kb = (lane >> 4) * 2;

    v2f afr[8];
    const float* ap = A + (size_t)(b * T_ + tt * 16 + m) * F_;
#pragma unroll
    for (int kk = 0; kk < 8; ++kk)
        afr[kk] = *(const v2f*)(ap + kk * 4 + kb);

#pragma unroll 1
    for (int nti = 0; nti < 8; ++nti) {
        const int n = (ng * 8 + nti) * 16 + m;
        v8f acc = {};
#pragma unroll
        for (int kk = 0; kk < 8; ++kk) {
            v2f bfr;
            bfr.x = u2[(size_t)(kk * 4 + kb) * N_ + n];
            bfr.y = u2[(size_t)(kk * 4 + kb + 1) * N_ + n];
            acc = wmma_f32_k4(afr[kk], bfr, acc);
        }
        float* dp = D + (size_t)(b * T_ + tt * 16 + (lane >> 4) * 8) * N_ + n;
#pragma unroll
        for (int v = 0; v < 8; ++v) dp[(size_t)v * N_] = acc[v];
    }
}

// ---------------------------------------------------------------------------
// Kernel 3: inner[b,t,s] = sum_n D[b,t,n] * C[b,n,s]  (per b: 288x2048 @ 2048x288)
// One wave per 32x32 output block (2x2 tiles of 16x16) -> frag reuse halves
// L2 traffic. 16*9*9 = 1296 waves, K-loop = 512 steps of 4, 4 WMMA each.
// ---------------------------------------------------------------------------
__global__ __launch_bounds__(128) void k3_gemm(
    const float* __restrict__ D, const float* __restrict__ C,
    float* __restrict__ inner) {
    const int wave = (blockIdx.x * 128 + threadIdx.x) >> 5;  // 1296 waves
    const int lane = threadIdx.x & 31;
    const int st = wave % 9;
    const int tt = (wave / 9) % 9;
    const int b = wave / 81;
    const int m = lane & 15;
    const int half = lane >> 4;
    const int kb = half * 2;
    const int t0 = tt * 32, s0 = st * 32;

    v8f acc00 = {}, acc01 = {}, acc10 = {}, acc11 = {};
    const float* dp0 = D + (size_t)(b * T_ + t0 + m) * N_;
    const float* dp1 = dp0 + (size_t)16 * N_;
    const float* cb = C + (size_t)b * N_ * T_;

    for (int k = 0; k < N_; k += 4) {
        v2f a0 = *(const v2f*)(dp0 + k + kb);
        v2f a1 = *(const v2f*)(dp1 + k + kb);
        const float* cr0 = cb + (size_t)(k + kb) * T_ + s0 + m;
        const float* cr1 = cr0 + T_;
        v2f b0, b1;
        b0.x = cr0[0];  b0.y = cr1[0];
        b1.x = cr0[16]; b1.y = cr1[16];
        acc00 = wmma_f32_k4(a0, b0, acc00);
        acc01 = wmma_f32_k4(a0, b1, acc01);
        acc10 = wmma_f32_k4(a1, b0, acc10);
        acc11 = wmma_f32_k4(a1, b1, acc11);
    }

    float* ip = inner + (size_t)(b * T_ + t0 + half * 8) * T_ + s0 + m;
#pragma unroll
    for (int v = 0; v < 8; ++v) {
        ip[(size_t)v * T_] = acc00[v];
        ip[(size_t)v * T_ + 16] = acc01[v];
        ip[(size_t)(v + 16) * T_] = acc10[v];
        ip[(size_t)(v + 16) * T_ + 16] = acc11[v];
    }
}

// ---------------------------------------------------------------------------
// Kernel 4: out[b,t,s] = softmax_s( vt[t,s] * sigmoid(inner[b,t,s] + bias[t,s]) )
// One wave32 per row of 288 (9 elems/lane); shuffle reductions for max & sum.
// ---------------------------------------------------------------------------
__global__ __launch_bounds__(128) void k4_softmax(
    const float* __restrict__ inner, const float* __restrict__ vt,
    const float* __restrict__ bias, float* __restrict__ out) {
    const int row = blockIdx.x * 4 + (threadIdx.x >> 5);  // 0 .. B_*T_-1
    const int lane = threadIdx.x & 31;
    const int t = row % T_;
    const float* ip = inner + (size_t)row * T_;
    const float* vp = vt + (size_t)t * T_;
    const float* bp = bias + (size_t)t * T_;

    float z[9];
    float mx = -3.402823466e38f;
#pragma unroll
    for (int j = 0; j < 9; ++j) {
        const int s = lane + j * 32;
        const float iv = ip[s] + bp[s];
        const float sg = 1.0f / (1.0f + __expf(-iv));
        z[j] = vp[s] * sg;
        mx = fmaxf(mx, z[j]);
    }
#pragma unroll
    for (int off = 16; off; off >>= 1) mx = fmaxf(mx, __shfl_xor(mx, off, 32));
    float sum = 0.f;
#pragma unroll
    for (int j = 0; j < 9; ++j) { z[j] = __expf(z[j] - mx); sum += z[j]; }
#pragma unroll
    for (int off = 16; off; off >>= 1) sum += __shfl_xor(sum, off, 32);
    const float inv = 1.0f / sum;
    float* op = out + (size_t)row * T_;
#pragma unroll
    for (int j = 0; j < 9; ++j) op[lane + j * 32] = z[j] * inv;
}

// ---------------------------------------------------------------------------
extern "C" void kernel_launch(void* const* d_in, const int* in_sizes, int n_in,
                              void* d_out, int out_size, void* d_ws, size_t ws_size,
                              hipStream_t stream) {
    (void)in_sizes; (void)n_in; (void)out_size; (void)ws_size;
    const float* x    = (const float*)d_in[0];
    const float* u1   = (const float*)d_in[1];
    const float* u2   = (const float*)d_in[2];
    const float* u3   = (const float*)d_in[3];
    const float* vt   = (const float*)d_in[4];
    const float* bias = (const float*)d_in[5];
    float* out = (float*)d_out;

    // Workspace layout (bytes). Apart (18.9MB) aliases D (37.7MB): Apart is
    // fully consumed by k1b before k2 writes D (stream-ordered).
    char* ws = (char*)d_ws;
    const size_t SZ_D     = (size_t)B_ * T_ * N_ * 4;   // 37,748,736 (also covers Apart)
    const size_t SZ_A     = (size_t)B_ * T_ * F_ * 4;   //    589,824
    const size_t SZ_C     = (size_t)B_ * N_ * T_ * 4;   // 37,748,736
    float* Apart = (float*)ws;
    float* D     = (float*)ws;
    float* A     = (float*)(ws + SZ_D);
    float* C     = (float*)(ws + SZ_D + SZ_A);
    float* inner = (float*)(ws + SZ_D + SZ_A + SZ_C);   // 5,308,416

    // 1) Stream x once (HBM-bound ~52us): fused dual reduction.
    k1_stream<<<dim3(NCHUNKS, B_), T_, 0, stream>>>(x, u1, u3, C, Apart);
    // 1b) Deterministic reduce of partial A.
    k1b_reduceA<<<(B_ * T_ * F_ + 255) / 256, 256, 0, stream>>>(Apart, A);
    // 2) D = A @ u2 per batch (f32 WMMA).
    k2_gemm<<<(B_ * 18 * 16) / 8, 256, 0, stream>>>(A, u2, D);
    // 3) inner = D @ C per batch (f32 WMMA, 2x2 tile).
    k3_gemm<<<(B_ * 9 * 9) / 4, 128, 0, stream>>>(D, C, inner);
    // 4) Fused sigmoid * vt + softmax.
    k4_softmax<<<(B_ * T_) / 4, 128, 0, stream>>>(inner, vt, bias, out);
}